// ResidualLFQ_45148696216337
// MI455X (gfx1250) — compile-verified
//
#include <hip/hip_runtime.h>
#include <hip/hip_bf16.h>

// ---------------------------------------------------------------------------
// ResidualLFQ for MI455X (gfx1250, wave32).
//
// Math restructuring: softmax over 16384 sign-codes factorizes into two
// independent 128-way softmaxes (high 7 bits / low 7 bits):
//   p[c] = pA[c>>7] * pB[c&127]
// so  avg_prob = (1/T) * A^T B  is a [128 x 4096] x [4096 x 128] f32 GEMM
// -> V_WMMA_F32_16X16X4_F32 (exact f32 accumulate).
// project_out is a [4096x16]x[16x1024] f32 GEMM -> also f32 WMMA.
// Everything else is per-token and fused into one kernel.
// ---------------------------------------------------------------------------

typedef float v2f __attribute__((ext_vector_type(2)));
typedef float v8f __attribute__((ext_vector_type(8)));

#define NTOK      4096          // 2*2048 tokens
#define CDIM      14
#define HSTRIDE   16            // padded token stride for h / qout
#define NUMQ      4
#define LEPS_F    (-11.512925464970229f)   // log(1e-5)

// workspace layout (floats)
#define WS_P      0                        // pA/pB transposed: [4][2][128][4096]
#define WS_H      (4*2*128*4096)           // h  [4096][16]
#define WS_QOUT   (WS_H   + NTOK*HSTRIDE)  // qout [4096][16]
#define WS_CMAT   (WS_QOUT+ NTOK*HSTRIDE)  // avg_prob [4][16384]
#define WS_SCAL   (WS_CMAT+ NUMQ*16384)    // [0..3]=sum H_t, [4..7]=sum commit_t

// d_out layout (4-byte elements)
#define OUT_F     (NTOK*1024)              // out  [2,2048,1024] f32
#define OUT_IDX   OUT_F                    // indices [2,2048,4] int32
#define OUT_LOSS  (OUT_F + NTOK*NUMQ)      // losses [4] f32

// ---------------------------------------------------------------- K0: init
__global__ void k0_init(float* __restrict__ scal) {
    if (threadIdx.x < 16) scal[threadIdx.x] = 0.0f;
}

// ------------------------------------------------- K1: h = x @ w_in + b_in
// one thread per (token, d) output; 16 tokens per 256-thread block
__global__ void k1_proj_in(const float* __restrict__ x,
                           const float* __restrict__ w_in,
                           const float* __restrict__ b_in,
                           float* __restrict__ H) {
    int tid   = threadIdx.x;
    int token = blockIdx.x * 16 + (tid >> 4);
    int d     = tid & 15;
    float acc = 0.0f;
    if (d < CDIM) {
        acc = b_in[d];
        const float* xr = x + (size_t)token * 1024;
        #pragma unroll 8
        for (int k = 0; k < 1024; ++k)
            acc = fmaf(xr[k], w_in[k * CDIM + d], acc);
    }
    H[token * HSTRIDE + d] = acc;
}

// --------------------------------------- K2: fused 4-layer LFQ, per token
// one wave per token (wave32); each lane owns 4 of 128 entries of pA and pB
__global__ void k2_lfq(const float* __restrict__ H,
                       float* __restrict__ P,
                       float* __restrict__ QOUT,
                       int*   __restrict__ IDX,
                       float* __restrict__ SCAL) {
    const int lane = threadIdx.x & 31;
    const int t    = blockIdx.x * 8 + (threadIdx.x >> 5);

    float h[CDIM], qsum[CDIM];
    #pragma unroll
    for (int d = 0; d < CDIM; ++d) { h[d] = H[t * HSTRIDE + d]; qsum[d] = 0.0f; }

    for (int l = 0; l < NUMQ; ++l) {
        const float scale = 1.0f / (float)(1 << l);
        const float a     = 200.0f * scale;          // 2 * INV_TEMP * scale

        // analytic softmax maxima: a * sum |h|
        float mA = 0.0f, mB = 0.0f;
        #pragma unroll
        for (int d = 0; d < 7;  ++d) mA += fabsf(h[d]);
        #pragma unroll
        for (int d = 7; d < 14; ++d) mB += fabsf(h[d]);
        mA *= a; mB *= a;

        float pA[4], lA[4], pB[4], lB[4];
        float sumA = 0.0f, sumB = 0.0f;
        #pragma unroll
        for (int r = 0; r < 4; ++r) {
            const int i = r * 32 + lane;             // code high/low 7-bit value
            float uA = 0.0f, uB = 0.0f;
            #pragma unroll
            for (int b = 0; b < 7; ++b) {
                // bit b of hi-part <-> dim (6-b); bit b of lo-part <-> dim (13-b)
                float vA = h[6  - b];
                float vB = h[13 - b];
                uA += ((i >> b) & 1) ? vA : -vA;
                uB += ((i >> b) & 1) ? vB : -vB;
            }
            uA = fmaf(a, uA, -mA);
            uB = fmaf(a, uB, -mB);
            float eA = expf(uA), eB = expf(uB);
            lA[r] = uA; pA[r] = eA; sumA += eA;
            lB[r] = uB; pB[r] = eB; sumB += eB;
        }
        #pragma unroll
        for (int off = 16; off > 0; off >>= 1) {
            sumA += __shfl_xor(sumA, off, 32);
            sumB += __shfl_xor(sumB, off, 32);
        }
        const float izA = 1.0f / sumA, lzA = logf(sumA);
        const float izB = 1.0f / sumB, lzB = logf(sumB);
        #pragma unroll
        for (int r = 0; r < 4; ++r) {
            pA[r] *= izA; lA[r] -= lzA;              // lA = log pA exactly
            pB[r] *= izB; lB[r] -= lzB;
        }

        // store pA/pB transposed: P[l][0][i][t], P[l][1][j][t]  (for WMMA GEMM)
        {
            float* Ab = P + (size_t)(l * 2 + 0) * 128 * 4096 + t;
            float* Bb = P + (size_t)(l * 2 + 1) * 128 * 4096 + t;
            #pragma unroll
            for (int r = 0; r < 4; ++r) {
                Ab[(size_t)(r * 32 + lane) * 4096] = pA[r];
                Bb[(size_t)(r * 32 + lane) * 4096] = pB[r];
            }
        }

        // clipped per-sample entropy: -sum_{i,j} pA_i pB_j max(lA_i+lB_j, log eps)
        float acc = 0.0f;
        #pragma unroll
        for (int r = 0; r < 4; ++r) {
            float pBr = pB[r], lBr = lB[r];
            for (int jl = 0; jl < 32; ++jl) {
                float pj = __shfl(pBr, jl, 32);
                float lj = __shfl(lBr, jl, 32);
                #pragma unroll
                for (int ri = 0; ri < 4; ++ri) {
                    float s = lA[ri] + lj;
                    acc = fmaf(pA[ri] * pj, fmaxf(s, LEPS_F), acc);
                }
            }
        }
        #pragma unroll
        for (int off = 16; off > 0; off >>= 1) acc += __shfl_xor(acc, off, 32);

        // quantize, commit, indices, residual update (identical on all lanes)
        float commit = 0.0f; int idx = 0;
        #pragma unroll
        for (int d = 0; d < CDIM; ++d) {
            bool  pos  = h[d] > 0.0f;
            float q    = pos ? scale : -scale;
            idx       |= (pos ? 1 : 0) << (13 - d);
            float diff = h[d] - q;
            commit    += diff * diff;
            h[d]       = diff;
            qsum[d]   += q;
        }
        if (lane == 0) {
            atomicAdd(&SCAL[l],     -acc);           // sum over tokens of H_t
            atomicAdd(&SCAL[4 + l], commit);
            IDX[t * NUMQ + l] = idx;
        }
    }

    // write padded quantized_out row
    float v = qsum[0];
    #pragma unroll
    for (int d = 1; d < CDIM; ++d) v = (lane == d) ? qsum[d] : v;
    if (lane < CDIM)                 QOUT[t * HSTRIDE + lane] = v;
    else if (lane < HSTRIDE)         QOUT[t * HSTRIDE + lane] = 0.0f;
}

// ---------------------- K3: avg_prob[l] = (1/T) * A^T B  via f32 WMMA
// one wave per 16x16 tile; 4 layers x 8x8 tiles = 256 waves; K = 4096
__global__ void k3_avgprob(const float* __restrict__ P, float* __restrict__ C) {
    const int gw   = (blockIdx.x * blockDim.x + threadIdx.x) >> 5;
    const int lane = threadIdx.x & 31;
    const int l    = gw >> 6;
    const int ti   = (gw >> 3) & 7;
    const int tj   = gw & 7;
    const int half = lane >> 4, lm = lane & 15;

    const float* At = P + (size_t)(l * 2 + 0) * 128 * 4096;
    const float* Bt = P + (size_t)(l * 2 + 1) * 128 * 4096;
    const float* ar = At + (size_t)(ti * 16 + lm) * 4096 + 2 * half; // A[m][k]: k contiguous
    const float* br = Bt + (size_t)(tj * 16 + lm) * 4096 + 2 * half; // B[k][n] rows

    v8f c0 = {}, c1 = {}, c2 = {}, c3 = {};   // 4 chains to pipeline the XDL
    for (int t = 0; t < 4096; t += 16) {
        v2f a0 = *(const v2f*)(ar + t +  0), b0 = *(const v2f*)(br + t +  0);
        v2f a1 = *(const v2f*)(ar + t +  4), b1 = *(const v2f*)(br + t +  4);
        v2f a2 = *(const v2f*)(ar + t +  8), b2 = *(const v2f*)(br + t +  8);
        v2f a3 = *(const v2f*)(ar + t + 12), b3 = *(const v2f*)(br + t + 12);
        c0 = __builtin_amdgcn_wmma_f32_16x16x4_f32(false, a0, false, b0, (short)0, c0, false, false);
        c1 = __builtin_amdgcn_wmma_f32_16x16x4_f32(false, a1, false, b1, (short)0, c1, false, false);
        c2 = __builtin_amdgcn_wmma_f32_16x16x4_f32(false, a2, false, b2, (short)0, c2, false, false);
        c3 = __builtin_amdgcn_wmma_f32_16x16x4_f32(false, a3, false, b3, (short)0, c3, false, false);
    }
    v8f acc = (c0 + c1) + (c2 + c3);
    const float inv = 1.0f / (float)NTOK;
    float* Cl = C + l * 16384;
    #pragma unroll
    for (int v = 0; v < 8; ++v) {
        int row = ti * 16 + v + 8 * half;            // C layout: VGPR v -> M=v / M=8+v
        Cl[row * 128 + tj * 16 + lm] = acc[v] * inv;
    }
}

// ------------- K4: out = qout @ w_out + b_out  via f32 WMMA (K padded to 16)
__global__ void k4_proj_out(const float* __restrict__ QOUT,
                            const float* __restrict__ w_out,
                            const float* __restrict__ b_out,
                            float* __restrict__ out) {
    const int gw   = (blockIdx.x * blockDim.x + threadIdx.x) >> 5; // 0..16383
    const int lane = threadIdx.x & 31;
    const int tm   = gw >> 6;                       // 256 M tiles
    const int tn   = gw & 63;                       // 64  N tiles
    const int m0   = tm * 16, n0 = tn * 16;
    const int half = lane >> 4, lm = lane & 15;

    const float* ar = QOUT + (size_t)(m0 + lm) * HSTRIDE + 2 * half;
    v8f acc = {};
    #pragma unroll
    for (int ks = 0; ks < 4; ++ks) {
        v2f a = *(const v2f*)(ar + 4 * ks);
        int kk = 4 * ks + 2 * half;
        v2f b;
        b[0] = (kk     < CDIM) ? w_out[(kk    ) * 1024 + n0 + lm] : 0.0f;
        b[1] = (kk + 1 < CDIM) ? w_out[(kk + 1) * 1024 + n0 + lm] : 0.0f;
        acc = __builtin_amdgcn_wmma_f32_16x16x4_f32(false, a, false, b, (short)0, acc, false, false);
    }
    const float bias = b_out[n0 + lm];
    #pragma unroll
    for (int v = 0; v < 8; ++v)
        out[(size_t)(m0 + v + 8 * half) * 1024 + n0 + lm] = acc[v] + bias;
}

// ------------------------------- K5: codebook entropy + loss assembly
__global__ void k5_losses(const float* __restrict__ C,
                          const float* __restrict__ SCAL,
                          float* __restrict__ losses) {
    const int l = blockIdx.x, tid = threadIdx.x;
    const float* Cl = C + l * 16384;
    float acc = 0.0f;
    for (int i = tid; i < 16384; i += 256) {
        float q = Cl[i];
        acc += q * fmaxf(logf(q), LEPS_F);           // q==0 -> 0 * LEPS = 0
    }
    __shared__ float red[256];
    red[tid] = acc; __syncthreads();
    for (int s = 128; s > 0; s >>= 1) {
        if (tid < s) red[tid] += red[tid + s];
        __syncthreads();
    }
    if (tid == 0) {
        float cb_ent = -red[0];
        float ps_ent = SCAL[l]     / (float)NTOK;
        float commit = SCAL[4 + l] / (float)(NTOK * CDIM);
        losses[l] = (ps_ent - 1.0f * cb_ent) * 0.1f + commit * 0.25f;
    }
}

// ---------------------------------------------------------------------------
extern "C" void kernel_launch(void* const* d_in, const int* in_sizes, int n_in,
                              void* d_out, int out_size, void* d_ws, size_t ws_size,
                              hipStream_t stream) {
    const float* x     = (const float*)d_in[0];
    const float* w_in  = (const float*)d_in[1];
    const float* b_in  = (const float*)d_in[2];
    const float* w_out = (const float*)d_in[3];
    const float* b_out = (const float*)d_in[4];

    float* out    = (float*)d_out;
    int*   idxout = (int*)((float*)d_out + OUT_IDX);
    float* losses = (float*)d_out + OUT_LOSS;

    float* W    = (float*)d_ws;          // needs ~17.6 MB
    float* P    = W + WS_P;
    float* H    = W + WS_H;
    float* QOUT = W + WS_QOUT;
    float* CMAT = W + WS_CMAT;
    float* SCAL = W + WS_SCAL;

    k0_init   <<<1,    32,  0, stream>>>(SCAL);
    k1_proj_in<<<256,  256, 0, stream>>>(x, w_in, b_in, H);
    k2_lfq    <<<512,  256, 0, stream>>>(H, P, QOUT, idxout, SCAL);
    k3_avgprob<<<32,   256, 0, stream>>>(P, CMAT);
    k4_proj_out<<<2048,256, 0, stream>>>(QOUT, w_out, b_out, out);
    k5_losses <<<4,    256, 0, stream>>>(CMAT, SCAL, losses);
}